// VisionMamba_47931835023568
// MI455X (gfx1250) — compile-verified
//
#include <hip/hip_runtime.h>
#include <hip/hip_bf16.h>

// ---------------------------------------------------------------------------
// VisionMamba forward for gfx1250 (MI455X).
// fp32 throughout; GEMMs use V_WMMA_F32_16X16X4_F32 (exact fp32 matrix math).
// All GEMM dims padded to 16-multiples -> branch-free inner loops.
// ---------------------------------------------------------------------------

#define D_MODEL 192
#define D_INNER 384
#define D_STATE 16
#define DT_RANK 12
#define DEPTH   24
#define BATCH   8
#define SEQ     197                       // 196 patches + cls at position 98
#define NPATCH  196
#define ROWS    (BATCH * SEQ)             // 1576
#define ROWS_P  1584                      // 99 * 16 (padded row count)
#define XD_W    (DT_RANK + 2 * D_STATE)   // 44
#define XD_P    48                        // padded x_proj output width

typedef float v2f __attribute__((ext_vector_type(2)));
typedef float v8f __attribute__((ext_vector_type(8)));

// ---------------------------------------------------------------------------
// Wave-level fp32 WMMA GEMM:  Out[Mp,N] = A[Mp,K] @ W[N,K]^T (+ bias[N])
// Mp, N, K are all multiples of 16 -> no bounds checks anywhere.
// Each wave (blockDim=32) computes a 16 x (16*NT) tile; K unrolled by 4
// (16 K-elements = 4*NT WMMAs per iteration, loads batched up front).
// grid.x = Mp/16, grid.y = N/(16*NT).
// Fragment layouts per CDNA5 ISA 7.12.2 (32-bit A 16x4, 32-bit C/D 16x16).
// ---------------------------------------------------------------------------
template <int NT>
__global__ __launch_bounds__(32) void gemm_f32_wmma(
    const float* __restrict__ A, const float* __restrict__ W,
    const float* __restrict__ bias, float* __restrict__ Out,
    int N, int K)
{
    const int lane = threadIdx.x & 31;
    const int l16  = lane & 15;
    const int half = lane >> 4;            // 0: lanes 0-15, 1: lanes 16-31
    const int mt   = blockIdx.x;
    const int nbase = blockIdx.y * (16 * NT);

    // Per-lane base pointers (lane-half owns K pair {0,1} or {2,3} of each step)
    const float* Ap = A + (size_t)(mt * 16 + l16) * K + (half << 1);
    const float* Wp[NT];
#pragma unroll
    for (int t = 0; t < NT; ++t)
        Wp[t] = W + (size_t)(nbase + t * 16 + l16) * K + (half << 1);

    v8f acc[NT];
#pragma unroll
    for (int t = 0; t < NT; ++t) acc[t] = (v8f)0.0f;

    for (int k0 = 0; k0 < K; k0 += 16) {
        v2f a[4];
#pragma unroll
        for (int u = 0; u < 4; ++u)
            a[u] = *(const v2f*)(Ap + k0 + 4 * u);
        v2f b[NT][4];
#pragma unroll
        for (int t = 0; t < NT; ++t)
#pragma unroll
            for (int u = 0; u < 4; ++u)
                b[t][u] = *(const v2f*)(Wp[t] + k0 + 4 * u);
#pragma unroll
        for (int u = 0; u < 4; ++u)
#pragma unroll
            for (int t = 0; t < NT; ++t)
                acc[t] = __builtin_amdgcn_wmma_f32_16x16x4_f32(
                    false, a[u], false, b[t][u], (short)0, acc[t], false, false);
    }

#pragma unroll
    for (int t = 0; t < NT; ++t) {
        int col = nbase + t * 16 + l16;
        float bv = bias ? bias[col] : 0.0f;
#pragma unroll
        for (int v = 0; v < 8; ++v) {
            int row = mt * 16 + v + (half << 3);   // lanes>=16 hold M = v+8
            Out[(size_t)row * N + col] = acc[t][v] + bv;
        }
    }
}

// ---------------------------------------------------------------------------
// Zero-pad x_proj_w (DEPTH,44,384) -> (DEPTH,48,384) so the x_proj GEMM is
// branch-free (rows 44..47 are zeros).
// ---------------------------------------------------------------------------
__global__ __launch_bounds__(256) void prepack_xpw_kernel(
    const float* __restrict__ src, float* __restrict__ dst)
{
    int tid = blockIdx.x * 256 + threadIdx.x;
    if (tid >= DEPTH * XD_P * D_INNER) return;
    int k = tid % D_INNER;
    int r = (tid / D_INNER) % XD_P;
    int l = tid / (D_INNER * XD_P);
    dst[tid] = (r < XD_W) ? src[((size_t)l * XD_W + r) * D_INNER + k] : 0.0f;
}

// ---------------------------------------------------------------------------
// Patchify: x(8,3,224,224) -> p(8,196,768) row-major (im2col for 16x16/s16)
// p[b, ph*14+pw, c*256 + i*16 + j] = x[b, c, ph*16+i, pw*16+j]
// ---------------------------------------------------------------------------
__global__ __launch_bounds__(256) void patchify_kernel(
    const float* __restrict__ x, float* __restrict__ p)
{
    int tid = blockIdx.x * 256 + threadIdx.x;
    if (tid >= BATCH * NPATCH * 768) return;
    int k  = tid % 768;
    int pi = (tid / 768) % NPATCH;
    int b  = tid / (768 * NPATCH);
    int c = k >> 8, rem = k & 255, i = rem >> 4, j = rem & 15;
    int ph = pi / 14, pw = pi % 14;
    p[tid] = x[(((size_t)b * 3 + c) * 224 + (ph * 16 + i)) * 224 + (pw * 16 + j)];
}

// ---------------------------------------------------------------------------
// Assemble token sequence: insert cls at l=98, add pos_embed -> h(8,197,192)
// ---------------------------------------------------------------------------
__global__ __launch_bounds__(256) void assemble_kernel(
    const float* __restrict__ tok, const float* __restrict__ cls,
    const float* __restrict__ pos, float* __restrict__ h)
{
    int tid = blockIdx.x * 256 + threadIdx.x;
    if (tid >= BATCH * SEQ * D_MODEL) return;
    int c = tid % D_MODEL;
    int l = (tid / D_MODEL) % SEQ;
    int b = tid / (D_MODEL * SEQ);
    float v;
    if (l < 98)       v = tok[((size_t)b * NPATCH + l) * D_MODEL + c];
    else if (l == 98) v = cls[c];
    else              v = tok[((size_t)b * NPATCH + (l - 1)) * D_MODEL + c];
    h[tid] = v + pos[l * D_MODEL + c];
}

// ---------------------------------------------------------------------------
// res = (first ? h : res + h); hn = rmsnorm(res) * w.  One wave per row.
// ---------------------------------------------------------------------------
__global__ __launch_bounds__(32) void resnorm_kernel(
    const float* __restrict__ h, float* __restrict__ res,
    float* __restrict__ hn, const float* __restrict__ w, int first)
{
    int row = blockIdx.x, lane = threadIdx.x;
    const float* hp = h + (size_t)row * D_MODEL;
    float* rp = res + (size_t)row * D_MODEL;
    float* op = hn + (size_t)row * D_MODEL;
    float v[6], ss = 0.0f;
#pragma unroll
    for (int j = 0; j < 6; ++j) {
        int c = j * 32 + lane;
        float x = hp[c];
        if (!first) x += rp[c];
        v[j] = x; ss += x * x;
    }
#pragma unroll
    for (int off = 16; off >= 1; off >>= 1) ss += __shfl_xor(ss, off, 32);
    float sc = rsqrtf(ss * (1.0f / D_MODEL) + 1e-5f);
#pragma unroll
    for (int j = 0; j < 6; ++j) {
        int c = j * 32 + lane;
        rp[c] = v[j];
        op[c] = v[j] * sc * w[c];
    }
}

// ---------------------------------------------------------------------------
// Depthwise causal conv1d (K=4) + SiLU over first half of xz -> xi
// ---------------------------------------------------------------------------
__global__ __launch_bounds__(256) void conv_silu_kernel(
    const float* __restrict__ xz, const float* __restrict__ cw,
    const float* __restrict__ cb, float* __restrict__ xi)
{
    int tid = blockIdx.x * 256 + threadIdx.x;
    if (tid >= ROWS * D_INNER) return;
    int c = tid % D_INNER;
    int l = (tid / D_INNER) % SEQ;
    int b = tid / (D_INNER * SEQ);
    const float* base = xz + (size_t)b * SEQ * (2 * D_INNER);
    float s = cb[c];
#pragma unroll
    for (int t = 0; t < 4; ++t) {
        int ls = l - 3 + t;
        if (ls >= 0) s += cw[c * 4 + t] * base[(size_t)ls * (2 * D_INNER) + c];
    }
    xi[(size_t)(b * SEQ + l) * D_INNER + c] = s / (1.0f + __expf(-s));   // SiLU
}

// ---------------------------------------------------------------------------
// delta = softplus(dt @ dtw^T + dtb);  dt = xd[:, :12]  (xd stride = 48)
// ---------------------------------------------------------------------------
__global__ __launch_bounds__(256) void delta_kernel(
    const float* __restrict__ xd, const float* __restrict__ dtw,
    const float* __restrict__ dtb, float* __restrict__ delta)
{
    int tid = blockIdx.x * 256 + threadIdx.x;
    if (tid >= ROWS * D_INNER) return;
    int d = tid % D_INNER;
    int r = tid / D_INNER;
    const float* row = xd + (size_t)r * XD_P;
    float s = dtb[d];
#pragma unroll
    for (int j = 0; j < DT_RANK; ++j) s += row[j] * dtw[d * DT_RANK + j];
    delta[tid] = (s > 20.0f) ? s : log1pf(__expf(s));
}

// ---------------------------------------------------------------------------
// Selective scan: one thread per (b, d); 16-entry state in registers.
// B/C (shared per (b,l)) staged in LDS by the first wave each step.
// y = (scan + u*Dp) * silu(z)
// ---------------------------------------------------------------------------
__global__ __launch_bounds__(128) void scan_kernel(
    const float* __restrict__ delta, const float* __restrict__ xi,
    const float* __restrict__ xd, const float* __restrict__ xz,
    const float* __restrict__ A_log, const float* __restrict__ Dp,
    float* __restrict__ y)
{
    __shared__ float bc[32];     // [0..15]=B, [16..31]=C for current (b,l)
    int b = blockIdx.x;
    int d = blockIdx.y * 128 + threadIdx.x;
    float Arow[D_STATE], h[D_STATE];
#pragma unroll
    for (int n = 0; n < D_STATE; ++n) {
        Arow[n] = -__expf(A_log[(size_t)d * D_STATE + n]);
        h[n] = 0.0f;
    }
    const float Dv = Dp[d];
    for (int l = 0; l < SEQ; ++l) {
        int row = b * SEQ + l;
        if (threadIdx.x < 32)
            bc[threadIdx.x] = xd[(size_t)row * XD_P + DT_RANK + threadIdx.x];
        __syncthreads();
        float dt = delta[(size_t)row * D_INNER + d];
        float u  = xi[(size_t)row * D_INNER + d];
        float du = dt * u, acc = 0.0f;
#pragma unroll
        for (int n = 0; n < D_STATE; ++n) {
            float hn = __expf(dt * Arow[n]) * h[n] + du * bc[n];
            h[n] = hn;
            acc += hn * bc[16 + n];
        }
        float z = xz[(size_t)row * (2 * D_INNER) + D_INNER + d];
        y[(size_t)row * D_INNER + d] = (acc + u * Dv) * (z / (1.0f + __expf(-z)));
        __syncthreads();
    }
}

// ---------------------------------------------------------------------------
// Final: rmsnorm(h + res) at l=98 only, then head matmul (192 -> 1000).
// One block per batch element.
// ---------------------------------------------------------------------------
__global__ __launch_bounds__(256) void head_kernel(
    const float* __restrict__ h, const float* __restrict__ res,
    const float* __restrict__ wn, const float* __restrict__ hw,
    const float* __restrict__ hb, float* __restrict__ out)
{
    __shared__ float v[D_MODEL];
    __shared__ float red[256];
    int b = blockIdx.x, t = threadIdx.x;
    size_t base = ((size_t)b * SEQ + 98) * D_MODEL;
    float x = 0.0f;
    if (t < D_MODEL) { x = h[base + t] + res[base + t]; v[t] = x; }
    red[t] = x * x;
    __syncthreads();
    for (int off = 128; off >= 1; off >>= 1) {
        if (t < off) red[t] += red[t + off];
        __syncthreads();
    }
    float sc = rsqrtf(red[0] * (1.0f / D_MODEL) + 1e-5f);
    if (t < D_MODEL) v[t] = v[t] * sc * wn[t];
    __syncthreads();
    for (int o = t; o < 1000; o += 256) {
        const float* wr = hw + (size_t)o * D_MODEL;
        float s = hb[o];
#pragma unroll 4
        for (int k = 0; k < D_MODEL; ++k) s += v[k] * wr[k];
        out[(size_t)b * 1000 + o] = s;
    }
}

// ---------------------------------------------------------------------------
// Host orchestration (all launches on `stream`; scratch from d_ws)
// ---------------------------------------------------------------------------
extern "C" void kernel_launch(void* const* d_in, const int* in_sizes, int n_in,
                              void* d_out, int out_size, void* d_ws, size_t ws_size,
                              hipStream_t stream) {
    (void)in_sizes; (void)n_in; (void)out_size; (void)ws_size;

    const float* x       = (const float*)d_in[0];
    const float* patch_w = (const float*)d_in[1];
    const float* patch_b = (const float*)d_in[2];
    const float* cls     = (const float*)d_in[3];
    const float* pos     = (const float*)d_in[4];
    const float* norm_w  = (const float*)d_in[5];
    const float* in_w    = (const float*)d_in[6];
    const float* conv_w  = (const float*)d_in[7];
    const float* conv_b  = (const float*)d_in[8];
    const float* xp_w    = (const float*)d_in[9];
    const float* dt_w    = (const float*)d_in[10];
    const float* dt_b    = (const float*)d_in[11];
    const float* A_log   = (const float*)d_in[12];
    const float* D_p     = (const float*)d_in[13];
    const float* out_w   = (const float*)d_in[14];
    const float* normf_w = (const float*)d_in[15];
    const float* head_w  = (const float*)d_in[16];
    const float* head_b  = (const float*)d_in[17];
    float* out = (float*)d_out;

    // Workspace layout (floats); padded rows (1576->1584) hold garbage that
    // is produced/consumed only by discarded GEMM tile rows. Total ~24 MB.
    float* ws      = (float*)d_ws;
    float* pbuf    = ws;                    // 1568*768 = 1,204,224
    float* tok     = pbuf    + 1204224;     // 1568*192 =   301,056
    float* hbuf    = tok     + 301056;      // 1584*192 =   304,128
    float* resb    = hbuf    + 304128;      // 1576*192 =   302,592
    float* hnb     = resb    + 302592;      // 1584*192 =   304,128
    float* xzb     = hnb     + 304128;      // 1584*768 = 1,216,512
    float* xib     = xzb     + 1216512;     // 1584*384 =   608,256
    float* xdb     = xib     + 608256;      // 1584*48  =    76,032
    float* dltb    = xdb     + 76032;       // 1576*384 =   605,184
    float* yb      = dltb    + 605184;      // 1584*384 =   608,256
    float* xpw_pad = yb      + 608256;      // 24*48*384 =  442,368

    // One-time (per launch) weight prepack: zero-padded x_proj_w
    prepack_xpw_kernel<<<(DEPTH * XD_P * D_INNER + 255) / 256, 256, 0, stream>>>(
        xp_w, xpw_pad);

    // Patch embedding: im2col + WMMA GEMM (1568x192 = 1568x768 @ (192x768)^T)
    patchify_kernel<<<4704, 256, 0, stream>>>(x, pbuf);
    gemm_f32_wmma<4><<<dim3(98, 3), 32, 0, stream>>>(
        pbuf, patch_w, patch_b, tok, D_MODEL, 768);
    assemble_kernel<<<1182, 256, 0, stream>>>(tok, cls, pos, hbuf);

    for (int i = 0; i < DEPTH; ++i) {
        resnorm_kernel<<<ROWS, 32, 0, stream>>>(
            hbuf, resb, hnb, norm_w + (size_t)i * D_MODEL, i == 0);
        // in_proj: (1584,192) @ (768,192)^T -> xz (1584,768)
        gemm_f32_wmma<4><<<dim3(99, 12), 32, 0, stream>>>(
            hnb, in_w + (size_t)i * 2 * D_INNER * D_MODEL, nullptr, xzb,
            2 * D_INNER, D_MODEL);
        conv_silu_kernel<<<2364, 256, 0, stream>>>(
            xzb, conv_w + (size_t)i * D_INNER * 4, conv_b + (size_t)i * D_INNER, xib);
        // x_proj: (1584,384) @ (48,384)^T -> xd (1584,48)   [NT=3 covers N=48]
        gemm_f32_wmma<3><<<dim3(99, 1), 32, 0, stream>>>(
            xib, xpw_pad + (size_t)i * XD_P * D_INNER, nullptr, xdb,
            XD_P, D_INNER);
        delta_kernel<<<2364, 256, 0, stream>>>(
            xdb, dt_w + (size_t)i * D_INNER * DT_RANK, dt_b + (size_t)i * D_INNER, dltb);
        scan_kernel<<<dim3(BATCH, 3), 128, 0, stream>>>(
            dltb, xib, xdb, xzb,
            A_log + (size_t)i * D_INNER * D_STATE, D_p + (size_t)i * D_INNER, yb);
        // out_proj: (1584,384) @ (192,384)^T -> h (1584,192)
        gemm_f32_wmma<4><<<dim3(99, 3), 32, 0, stream>>>(
            yb, out_w + (size_t)i * D_MODEL * D_INNER, nullptr, hbuf,
            D_MODEL, D_INNER);
    }

    head_kernel<<<BATCH, 256, 0, stream>>>(hbuf, resb, normf_w, head_w, head_b, out);
}